// AspectFusionLayer_66219805770057
// MI455X (gfx1250) — compile-verified
//
#include <hip/hip_runtime.h>
#include <hip/hip_bf16.h>

#define B_ 4
#define N_ 512
#define D_ 128
#define NEG_SLOPE 0.2f
#define LN_EPS 1e-5f

typedef __attribute__((ext_vector_type(2))) float v2f;
typedef __attribute__((ext_vector_type(8))) float v8f;

// ---------------------------------------------------------------------------
// fast tanh: gfx1250 has a hardware V_TANH_F32 (TRANS pipe). Use the builtin
// when clang declares it; otherwise one v_exp_f32 + div fallback.
// ---------------------------------------------------------------------------
__device__ __forceinline__ float fast_tanh(float x) {
#if __has_builtin(__builtin_amdgcn_tanhf)
  return __builtin_amdgcn_tanhf(x);
#else
  float ax = fabsf(x);
  float t  = __expf(-2.0f * ax);          // v_exp_f32
  float r  = (1.0f - t) / (1.0f + t);     // in (0,1]; exp underflow -> r = 1
  return copysignf(r, x);
#endif
}

// ---------------------------------------------------------------------------
// Kernel 1: q = x @ Wq^T + bq ; k = x @ Wk^T + bk
// x: [B*N, D], W: [e, d] row-major (einsum "bnd,ed->bne")
// One block = 16 output rows; 8 waves, wave w owns columns [16w, 16w+16).
// Each wave: 32 k-steps, 2 WMMA f32 16x16x4 per step (q and k share A-frag).
// ---------------------------------------------------------------------------
__global__ __launch_bounds__(256) void proj_qk_kernel(
    const float* __restrict__ x,
    const float* __restrict__ Wq, const float* __restrict__ qb,
    const float* __restrict__ Wk, const float* __restrict__ kb,
    float* __restrict__ q, float* __restrict__ k) {
  const int tid  = threadIdx.x;
  const int wave = tid >> 5;
  const int lane = tid & 31;
  const int lo   = lane & 15;   // M (A,C) / N (B,C) index
  const int hi   = lane >> 4;   // selects K pair / M+8 half
  const int row0 = blockIdx.x * 16;
  const int e0   = wave * 16;

  v8f cq = {};
  v8f ck = {};
  const float* arow  = x  + (size_t)(row0 + lo) * D_;   // A: m = lo
  const float* wqrow = Wq + (size_t)(e0 + lo) * D_;     // B[k,n] = W[e0+n, k]
  const float* wkrow = Wk + (size_t)(e0 + lo) * D_;

  for (int ks = 0; ks < D_ / 4; ++ks) {
    const int kk = 4 * ks + 2 * hi;   // this lane's K base (components +0,+1)
    v2f a, bq, bk;
    a.x  = arow[kk];      a.y  = arow[kk + 1];
    bq.x = wqrow[kk];     bq.y = wqrow[kk + 1];
    bk.x = wkrow[kk];     bk.y = wkrow[kk + 1];
    cq = __builtin_amdgcn_wmma_f32_16x16x4_f32(false, a, false, bq,
                                               (short)0, cq, false, false);
    ck = __builtin_amdgcn_wmma_f32_16x16x4_f32(false, a, false, bk,
                                               (short)0, ck, false, false);
  }

  const float biasq = qb[e0 + lo];
  const float biask = kb[e0 + lo];
#pragma unroll
  for (int r = 0; r < 8; ++r) {
    const int m = r + 8 * hi;   // C layout: VGPR r -> M=r (lanes 0-15), M=r+8
    q[(size_t)(row0 + m) * D_ + e0 + lo] = cq[r] + biasq;
    k[(size_t)(row0 + m) * D_ + e0 + lo] = ck[r] + biask;
  }
}

// ---------------------------------------------------------------------------
// Kernel 2: fused scores -> leaky_relu -> softmax -> alpha@x -> +x -> LN
// One block per (batch, 16-row tile): grid = B * N/16 = 128 blocks, 256 thr.
// Never materializes t[B,N,N,D]; the e tile lives in LDS.
// ---------------------------------------------------------------------------
__global__ __launch_bounds__(256) void attn_fused_kernel(
    const float* __restrict__ x,
    const float* __restrict__ q, const float* __restrict__ k,
    const float* __restrict__ attn_w, const float* __restrict__ attn_b,
    const float* __restrict__ ln_g, const float* __restrict__ ln_b,
    float* __restrict__ out) {
  // rows padded by 4 floats: kills bank conflicts AND keeps rows 16B-aligned
  // (132*4 = 528 bytes = 33*16) so float4 LDS loads are legal.
  __shared__ __align__(16) float q_s[16][D_ + 4];
  __shared__ __align__(16) float w_s[D_];
  __shared__ __align__(16) float e_s[16][N_ + 4];
  __shared__ __align__(16) float y_s[16][D_ + 4];
  __shared__ float red_s[16][16];
  __shared__ float mu_s[16], rs_s[16];

  const int tid  = threadIdx.x;
  const int b    = blockIdx.x >> 5;          // N_/16 = 32 tiles per batch
  const int it   = blockIdx.x & 31;
  const int row0 = it * 16;
  const float* xb = x + (size_t)b * N_ * D_;
  const float* qp = q + ((size_t)b * N_ + row0) * D_;
  const float* kp = k + (size_t)b * N_ * D_;

  // ---- stage q tile + attn_w into LDS ----
  for (int u = tid; u < 16 * D_; u += 256) q_s[u / D_][u % D_] = qp[u];
  if (tid < D_) w_s[tid] = attn_w[tid];
  __syncthreads();

  // ---- scores: e[i][j] = leaky(sum_d w_d*tanh(q[i,d]+k[j,d]) + ab) ----
  // j blocked by 4 (reuse q/w LDS reads), d vectorized by 4 (b128 loads).
  // Per 16 tanh: 4 global_load_b128 + 2 ds_load_b128 + 16 add/tanh/fma.
  const float ab = attn_b[0];
  const int   i  = tid & 15;     // row in tile
  const int   jb = tid >> 4;     // 0..15 -> contiguous j range of 32
  for (int jj = 0; jj < 32; jj += 4) {
    const int j = jb * 32 + jj;
    const float4* k0 = (const float4*)(kp + (size_t)(j + 0) * D_);
    const float4* k1 = (const float4*)(kp + (size_t)(j + 1) * D_);
    const float4* k2 = (const float4*)(kp + (size_t)(j + 2) * D_);
    const float4* k3 = (const float4*)(kp + (size_t)(j + 3) * D_);
    float a0 = 0.f, a1 = 0.f, a2 = 0.f, a3 = 0.f;
#pragma unroll 4
    for (int dq = 0; dq < D_ / 4; ++dq) {
      const float4 qv = *(const float4*)&q_s[i][4 * dq];
      const float4 wv = *(const float4*)&w_s[4 * dq];
      const float4 v0 = k0[dq], v1 = k1[dq], v2 = k2[dq], v3 = k3[dq];
      a0 = fmaf(wv.x, fast_tanh(qv.x + v0.x), a0);
      a0 = fmaf(wv.y, fast_tanh(qv.y + v0.y), a0);
      a0 = fmaf(wv.z, fast_tanh(qv.z + v0.z), a0);
      a0 = fmaf(wv.w, fast_tanh(qv.w + v0.w), a0);
      a1 = fmaf(wv.x, fast_tanh(qv.x + v1.x), a1);
      a1 = fmaf(wv.y, fast_tanh(qv.y + v1.y), a1);
      a1 = fmaf(wv.z, fast_tanh(qv.z + v1.z), a1);
      a1 = fmaf(wv.w, fast_tanh(qv.w + v1.w), a1);
      a2 = fmaf(wv.x, fast_tanh(qv.x + v2.x), a2);
      a2 = fmaf(wv.y, fast_tanh(qv.y + v2.y), a2);
      a2 = fmaf(wv.z, fast_tanh(qv.z + v2.z), a2);
      a2 = fmaf(wv.w, fast_tanh(qv.w + v2.w), a2);
      a3 = fmaf(wv.x, fast_tanh(qv.x + v3.x), a3);
      a3 = fmaf(wv.y, fast_tanh(qv.y + v3.y), a3);
      a3 = fmaf(wv.z, fast_tanh(qv.z + v3.z), a3);
      a3 = fmaf(wv.w, fast_tanh(qv.w + v3.w), a3);
    }
    float e0 = a0 + ab, e1 = a1 + ab, e2 = a2 + ab, e3 = a3 + ab;
    e_s[i][j + 0] = (e0 >= 0.f) ? e0 : NEG_SLOPE * e0;
    e_s[i][j + 1] = (e1 >= 0.f) ? e1 : NEG_SLOPE * e1;
    e_s[i][j + 2] = (e2 >= 0.f) ? e2 : NEG_SLOPE * e2;
    e_s[i][j + 3] = (e3 >= 0.f) ? e3 : NEG_SLOPE * e3;
  }
  __syncthreads();

  // ---- row softmax over j (16-thread row teams, LDS tree) ----
  float m = -INFINITY;
  for (int jj = 0; jj < 32; ++jj) m = fmaxf(m, e_s[i][jb * 32 + jj]);
  red_s[i][jb] = m;
  __syncthreads();
  float rowm = red_s[i][0];
#pragma unroll
  for (int t = 1; t < 16; ++t) rowm = fmaxf(rowm, red_s[i][t]);  // redundant, cheap
  __syncthreads();
  float s = 0.f;
  for (int jj = 0; jj < 32; ++jj) {
    const int j = jb * 32 + jj;
    const float ex = __expf(e_s[i][j] - rowm);
    e_s[i][j] = ex;
    s += ex;
  }
  red_s[i][jb] = s;
  __syncthreads();
  float rows = red_s[i][0];
#pragma unroll
  for (int t = 1; t < 16; ++t) rows += red_s[i][t];
  const float inv = 1.f / rows;
  for (int jj = 0; jj < 32; ++jj) e_s[i][jb * 32 + jj] *= inv;
  __syncthreads();

  // ---- out[i, d] = sum_j alpha[i,j] * x[j,d] via WMMA f32 16x16x4 ----
  // 8 waves; wave w owns the 16-wide d-slice [16w, 16w+16).
  const int wave = tid >> 5;
  const int lane = tid & 31;
  const int lo   = lane & 15;
  const int hi   = lane >> 4;
  const int d0   = wave * 16;
  v8f c = {};
  for (int jc = 0; jc < N_ / 4; ++jc) {
    const int kk = 4 * jc + 2 * hi;
    v2f a, bf;
    a.x  = e_s[lo][kk];                           // A[m=lo, k]
    a.y  = e_s[lo][kk + 1];
    bf.x = xb[(size_t)(kk + 0) * D_ + d0 + lo];   // B[k, n=lo] = x[j=k, d0+n]
    bf.y = xb[(size_t)(kk + 1) * D_ + d0 + lo];
    c = __builtin_amdgcn_wmma_f32_16x16x4_f32(false, a, false, bf,
                                              (short)0, c, false, false);
  }
#pragma unroll
  for (int r = 0; r < 8; ++r) {
    const int mi = r + 8 * hi;
    y_s[mi][d0 + lo] = c[r] + xb[(size_t)(row0 + mi) * D_ + d0 + lo]; // +residual
  }
  __syncthreads();

  // ---- LayerNorm over d ----
  if (tid < 16) {
    float su = 0.f;
    for (int d = 0; d < D_; ++d) su += y_s[tid][d];
    const float mu = su * (1.0f / D_);
    float var = 0.f;
    for (int d = 0; d < D_; ++d) {
      const float t = y_s[tid][d] - mu;
      var += t * t;
    }
    mu_s[tid] = mu;
    rs_s[tid] = rsqrtf(var * (1.0f / D_) + LN_EPS);
  }
  __syncthreads();
  const int i2 = tid >> 4;
  const int db = (tid & 15) * 8;
  const float mu = mu_s[i2], rs = rs_s[i2];
#pragma unroll
  for (int u = 0; u < 8; ++u) {
    const int d = db + u;
    out[((size_t)b * N_ + row0 + i2) * D_ + d] =
        (y_s[i2][d] - mu) * rs * ln_g[d] + ln_b[d];
  }
}

// ---------------------------------------------------------------------------
extern "C" void kernel_launch(void* const* d_in, const int* in_sizes, int n_in,
                              void* d_out, int out_size, void* d_ws, size_t ws_size,
                              hipStream_t stream) {
  const float* x      = (const float*)d_in[0];
  const float* Wq_w   = (const float*)d_in[1];
  const float* Wq_b   = (const float*)d_in[2];
  const float* Wk_w   = (const float*)d_in[3];
  const float* Wk_b   = (const float*)d_in[4];
  const float* attn_w = (const float*)d_in[5];
  const float* attn_b = (const float*)d_in[6];
  const float* ln_g   = (const float*)d_in[7];
  const float* ln_b   = (const float*)d_in[8];
  float* out = (float*)d_out;

  float* q = (float*)d_ws;                       // [B*N, D]
  float* k = q + (size_t)B_ * N_ * D_;           // [B*N, D]  (2 MB total)

  const int tiles = B_ * N_ / 16;                // 128
  proj_qk_kernel<<<tiles, 256, 0, stream>>>(x, Wq_w, Wq_b, Wk_w, Wk_b, q, k);
  attn_fused_kernel<<<tiles, 256, 0, stream>>>(x, q, k, attn_w, attn_b,
                                               ln_g, ln_b, out);
}